// FlowMatchingPointNeXt_83468394430486
// MI455X (gfx1250) — compile-verified
//
#include <hip/hip_runtime.h>
#include <math.h>

// ============================ types / helpers ============================

typedef __attribute__((ext_vector_type(2))) float v2f;
typedef __attribute__((ext_vector_type(8))) float v8f;

#define THREADS 256

__device__ __forceinline__ float silu_f(float v) { return v / (1.f + __expf(-v)); }

__device__ __forceinline__ v8f wmma4(v2f a, v2f b, v8f c) {
  return __builtin_amdgcn_wmma_f32_16x16x4_f32(false, a, false, b, (short)0, c, false, false);
}

// ============================ kernels ============================

// Y[b][m][n] = sum_k W[m][k] * X[b][k][n] + bias[m]
// One wave computes a 16(M) x 32(N) output tile via V_WMMA_F32_16X16X4_F32:
// two f32 accumulators sharing each A fragment (A loaded once, used twice).
// A (16x4 f32): lanes 0-15 row M=r hold K=k..k+1; lanes 16-31 hold K=k+2..k+3.
// B (4x16 f32): lanes 0-15 col N=r hold K=k..k+1 (v0/v1); lanes 16-31 K=k+2..k+3.
// C/D (16x16 f32): VGPR v, lanes 0-15 -> row v, lanes 16-31 -> row v+8.
// Guarded paths use clamped-address loads + value selects (no EXEC branches),
// keeping EXEC all-ones across every WMMA as the ISA requires.
__global__ void k_gemm_wmma(const float* __restrict__ W, const float* __restrict__ bias,
                            const float* __restrict__ X, float* __restrict__ Y,
                            int M, int K, int N, long long xStride, long long yStride)
{
  const int b = blockIdx.z;
  const float* Xb = X + (long long)b * xStride;
  float*       Yb = Y + (long long)b * yStride;
  const int wave = threadIdx.x >> 5;
  const int lane = threadIdx.x & 31;
  const int half = lane >> 4;
  const int r    = lane & 15;
  const int tn = (blockIdx.x * (blockDim.x >> 5) + wave) << 5;  // 32 cols / wave
  const int tm = blockIdx.y << 4;                               // 16 rows / wave
  if (tn >= N || tm >= M) return;  // wave-uniform: EXEC stays all-ones

  const int row  = tm + r;
  const int col0 = tn + r;
  const int col1 = tn + 16 + r;
  const int kmain = K & ~3;
  v8f acc0 = {}, acc1 = {};

  const bool fullTile = (tm + 16 <= M) && (tn + 32 <= N);  // wave-uniform
  if (fullTile) {
    // -------- fast path: zero guards in the K main loop --------
    const float* Wp = W  + (long long)row * K + 2 * half;
    const float* X0 = Xb + (long long)(2 * half) * N + col0;
    const float* X1 = Xb + (long long)(2 * half) * N + col1;
    int k = 0;
    for (; k + 8 <= kmain; k += 8) {
      v2f a0;  a0.x  = Wp[0];  a0.y  = Wp[1];
      v2f b00; b00.x = X0[0];  b00.y = X0[N];
      v2f b10; b10.x = X1[0];  b10.y = X1[N];
      v2f a1;  a1.x  = Wp[4];  a1.y  = Wp[5];
      v2f b01; b01.x = X0[(long long)4 * N]; b01.y = X0[(long long)5 * N];
      v2f b11; b11.x = X1[(long long)4 * N]; b11.y = X1[(long long)5 * N];
      acc0 = wmma4(a0, b00, acc0);
      acc1 = wmma4(a0, b10, acc1);
      acc0 = wmma4(a1, b01, acc0);
      acc1 = wmma4(a1, b11, acc1);
      Wp += 8;
      X0 += (long long)8 * N;
      X1 += (long long)8 * N;
    }
    for (; k < kmain; k += 4) {
      v2f a;  a.x  = Wp[0]; a.y  = Wp[1];
      v2f b0; b0.x = X0[0]; b0.y = X0[N];
      v2f b1; b1.x = X1[0]; b1.y = X1[N];
      acc0 = wmma4(a, b0, acc0);
      acc1 = wmma4(a, b1, acc1);
      Wp += 4;
      X0 += (long long)4 * N;
      X1 += (long long)4 * N;
    }
    if (kmain < K) {  // odd-K tail: unconditional clamped loads, then select
      const int ka = kmain + 2 * half;
      const bool k0 = ka < K, k1 = ka + 1 < K;
      const long long a0i = (long long)row * K + (k0 ? ka : K - 1);
      const long long a1i = (long long)row * K + (k1 ? ka + 1 : K - 1);
      const long long r0  = (long long)(k0 ? ka : 0) * N;
      const long long r1  = (long long)(k1 ? ka + 1 : 0) * N;
      const float av0 = W[a0i],        av1 = W[a1i];
      const float b00 = Xb[r0 + col0], b01 = Xb[r1 + col0];
      const float b10 = Xb[r0 + col1], b11 = Xb[r1 + col1];
      v2f a, b0, b1;
      a.x  = k0 ? av0 : 0.f;  a.y  = k1 ? av1 : 0.f;
      b0.x = k0 ? b00 : 0.f;  b0.y = k1 ? b01 : 0.f;
      b1.x = k0 ? b10 : 0.f;  b1.y = k1 ? b11 : 0.f;
      acc0 = wmma4(a, b0, acc0);
      acc1 = wmma4(a, b1, acc1);
    }
#pragma unroll
    for (int v = 0; v < 8; ++v) {
      const int orow = tm + v + 8 * half;
      const float bv = bias ? bias[orow] : 0.f;
      Yb[(long long)orow * N + col0] = acc0[v] + bv;
      Yb[(long long)orow * N + col1] = acc1[v] + bv;
    }
  } else {
    // -------- edge path: clamped addresses + value selects, no EXEC branches --------
    const bool rowOK = row  < M;
    const bool c0OK  = col0 < N;
    const bool c1OK  = col1 < N;
    const int rowC = rowOK ? row  : M - 1;
    const int c0C  = c0OK  ? col0 : 0;
    const int c1C  = c1OK  ? col1 : 0;
    for (int k = 0; k < K; k += 4) {
      const int ka = k + 2 * half;
      const bool k0 = ka < K, k1 = ka + 1 < K;
      const int ka0 = k0 ? ka : 0;
      const int ka1 = k1 ? ka + 1 : 0;
      const float av0 = W[(long long)rowC * K + ka0];
      const float av1 = W[(long long)rowC * K + ka1];
      const float b00 = Xb[(long long)ka0 * N + c0C];
      const float b01 = Xb[(long long)ka1 * N + c0C];
      const float b10 = Xb[(long long)ka0 * N + c1C];
      const float b11 = Xb[(long long)ka1 * N + c1C];
      v2f a, b0, b1;
      a.x  = (rowOK && k0) ? av0 : 0.f;  a.y  = (rowOK && k1) ? av1 : 0.f;
      b0.x = (c0OK  && k0) ? b00 : 0.f;  b0.y = (c0OK  && k1) ? b01 : 0.f;
      b1.x = (c1OK  && k0) ? b10 : 0.f;  b1.y = (c1OK  && k1) ? b11 : 0.f;
      acc0 = wmma4(a, b0, acc0);
      acc1 = wmma4(a, b1, acc1);
    }
#pragma unroll
    for (int v = 0; v < 8; ++v) {
      const int orow = tm + v + 8 * half;
      if (orow < M) {
        const float bv = bias ? bias[orow] : 0.f;
        if (c0OK) Yb[(long long)orow * N + col0] = acc0[v] + bv;
        if (c1OK) Yb[(long long)orow * N + col1] = acc1[v] + bv;
      }
    }
  }
}

// In-place GroupNorm over x[B][C][Npts], groups along C. Optional fused SiLU.
__global__ void k_groupnorm(float* __restrict__ x, const float* __restrict__ gamma,
                            const float* __restrict__ beta, int C, long long Npts, int G, int doSilu)
{
  const int b = blockIdx.x, g = blockIdx.y;
  const int cpg = C / G;
  const long long cnt = (long long)cpg * Npts;
  float* base = x + ((long long)b * C + (long long)g * cpg) * Npts;
  __shared__ float ssum[THREADS], ssq[THREADS];
  float s = 0.f, q = 0.f;
  for (long long i = threadIdx.x; i < cnt; i += blockDim.x) { float v = base[i]; s += v; q += v * v; }
  ssum[threadIdx.x] = s; ssq[threadIdx.x] = q;
  __syncthreads();
  for (int o = THREADS / 2; o > 0; o >>= 1) {
    if ((int)threadIdx.x < o) { ssum[threadIdx.x] += ssum[threadIdx.x + o]; ssq[threadIdx.x] += ssq[threadIdx.x + o]; }
    __syncthreads();
  }
  const float mean = ssum[0] / (float)cnt;
  const float var  = ssq[0] / (float)cnt - mean * mean;
  const float inv  = rsqrtf(var + 1e-5f);
  for (long long i = threadIdx.x; i < cnt; i += blockDim.x) {
    const int c = g * cpg + (int)(i / Npts);
    float v = (base[i] - mean) * inv * gamma[c] + beta[c];
    if (doSilu) v = silu_f(v);
    base[i] = v;
  }
}

// out[b][o] = act( sum_i W[o][i]*in[b][i] + bias[o] ), tiny per-batch linears.
__global__ void k_small_linear(const float* __restrict__ W, const float* __restrict__ bias,
                               const float* __restrict__ in, float* __restrict__ out,
                               int I, int O, int doSilu)
{
  const int b = blockIdx.y;
  for (int o = blockIdx.x * blockDim.x + threadIdx.x; o < O; o += gridDim.x * blockDim.x) {
    float s = bias[o];
    const float* w  = W + (long long)o * I;
    const float* xi = in + (long long)b * I;
    for (int i = 0; i < I; ++i) s += w[i] * xi[i];
    if (doSilu) s = silu_f(s);
    out[(long long)b * O + o] = s;
  }
}

__global__ void k_timestep_embed(const float* __restrict__ t, float* __restrict__ out, int half)
{
  const int b = blockIdx.y;
  const int i = blockIdx.x * blockDim.x + threadIdx.x;
  if (i >= 2 * half) return;
  const int j = (i < half) ? i : i - half;
  const float f = __expf(-__logf(10000.f) * (float)j / (float)half);
  const float a = t[b] * f;
  out[(long long)b * 2 * half + i] = (i < half) ? __cosf(a) : __sinf(a);
}

__global__ void k_embed_sum(const float* __restrict__ species_emb, const float* __restrict__ type_emb,
                            const int* __restrict__ sidx, const int* __restrict__ tidx,
                            const float* __restrict__ hp, float* __restrict__ cg, int Dm)
{
  const int b = blockIdx.y;
  const int j = blockIdx.x * blockDim.x + threadIdx.x;
  if (j >= Dm) return;
  cg[(long long)b * Dm + j] = species_emb[(long long)sidx[b] * Dm + j]
                            + type_emb[(long long)tidx[b] * Dm + j]
                            + hp[(long long)b * Dm + j];
}

// x[B][N][3] -> xyz[B][3][N]
__global__ void k_transpose_xyz(const float* __restrict__ x, float* __restrict__ xyz, int N)
{
  const int b = blockIdx.z;
  const int c = blockIdx.y;
  for (int n = blockIdx.x * blockDim.x + threadIdx.x; n < N; n += gridDim.x * blockDim.x)
    xyz[((long long)b * 3 + c) * N + n] = x[((long long)b * N + n) * 3 + c];
}

// Farthest point sampling: one block per batch, sequential npoint rounds with block argmax.
__global__ void k_fps(const float* __restrict__ xyz, int N, int npoint,
                      int* __restrict__ outIdx, float* __restrict__ dist)
{
  const int b = blockIdx.x;
  const float* px = xyz + (long long)b * 3 * N;
  float* d = dist + (long long)b * N;
  __shared__ float sv[THREADS];
  __shared__ int   si[THREADS];
  __shared__ int   far;
  if (threadIdx.x == 0) far = 0;
  for (int i = threadIdx.x; i < N; i += blockDim.x) d[i] = 1e10f;
  __syncthreads();
  for (int s = 0; s < npoint; ++s) {
    const int f = far;
    if (threadIdx.x == 0) outIdx[(long long)b * npoint + s] = f;
    const float cx = px[f], cy = px[N + f], cz = px[2 * N + f];
    float bestv = -1.f; int besti = 0;
    for (int i = threadIdx.x; i < N; i += blockDim.x) {
      const float dx = px[i] - cx, dy = px[N + i] - cy, dz = px[2 * N + i] - cz;
      const float dd = dx * dx + dy * dy + dz * dz;
      float cur = d[i];
      if (dd < cur) { cur = dd; d[i] = cur; }
      if (cur > bestv) { bestv = cur; besti = i; }
    }
    sv[threadIdx.x] = bestv; si[threadIdx.x] = besti;
    __syncthreads();
    for (int o = THREADS / 2; o > 0; o >>= 1) {
      if ((int)threadIdx.x < o && sv[threadIdx.x + o] > sv[threadIdx.x]) {
        sv[threadIdx.x] = sv[threadIdx.x + o]; si[threadIdx.x] = si[threadIdx.x + o];
      }
      __syncthreads();
    }
    if (threadIdx.x == 0) far = si[0];
    __syncthreads();
  }
}

// new_xyz[b][c][s] = xyz[b][c][idx[b][s]]
__global__ void k_gather_xyz(const float* __restrict__ xyz, const int* __restrict__ idx,
                             int N, int S, float* __restrict__ out)
{
  const int b = blockIdx.z;
  const int c = blockIdx.y;
  const int s = blockIdx.x * blockDim.x + threadIdx.x;
  if (s >= S) return;
  out[((long long)b * 3 + c) * S + s] = xyz[((long long)b * 3 + c) * N + idx[(long long)b * S + s]];
}

// Ascending-index scan == sorted idx + take first K (reference semantics).
__global__ void k_query_ball(const float* __restrict__ xyz, const float* __restrict__ nxyz,
                             int N, int S, float r2, int K, int* __restrict__ gidx)
{
  const int b = blockIdx.y;
  const int s = blockIdx.x * blockDim.x + threadIdx.x;
  if (s >= S) return;
  const float* px = xyz + (long long)b * 3 * N;
  const float cx = nxyz[((long long)b * 3 + 0) * S + s];
  const float cy = nxyz[((long long)b * 3 + 1) * S + s];
  const float cz = nxyz[((long long)b * 3 + 2) * S + s];
  int* out = gidx + ((long long)b * S + s) * K;
  int cnt = 0, first = 0; bool have = false;
  for (int i = 0; i < N && cnt < K; ++i) {
    const float dx = px[i] - cx, dy = px[N + i] - cy, dz = px[2 * N + i] - cz;
    if (dx * dx + dy * dy + dz * dz <= r2) { if (!have) { first = i; have = true; } out[cnt++] = i; }
  }
  const int pad = have ? first : 0;
  for (; cnt < K; ++cnt) out[cnt] = pad;
}

// g[b][c][s][k]: c<3 -> centered xyz, c>=3 -> gathered features.
__global__ void k_build_group(const float* __restrict__ xyz, const float* __restrict__ nxyz,
                              const float* __restrict__ feat, const int* __restrict__ gidx,
                              int N, int S, int K, int Cin, long long total, float* __restrict__ g)
{
  const int C = Cin + 3;
  for (long long t = blockIdx.x * (long long)blockDim.x + threadIdx.x; t < total;
       t += (long long)gridDim.x * blockDim.x) {
    const int k = (int)(t % K); long long r = t / K;
    const int s = (int)(r % S); r /= S;
    const int c = (int)(r % C); const int b = (int)(r / C);
    const int idx = gidx[((long long)b * S + s) * K + k];
    float v;
    if (c < 3) v = xyz[((long long)b * 3 + c) * N + idx] - nxyz[((long long)b * 3 + c) * S + s];
    else       v = feat[((long long)b * Cin + (c - 3)) * N + idx];
    g[t] = v;
  }
}

// out[t] = max_k h[t*K + k]   (h is [B][C][S][K] flattened)
__global__ void k_maxpool(const float* __restrict__ h, float* __restrict__ out, int K, long long total)
{
  for (long long t = blockIdx.x * (long long)blockDim.x + threadIdx.x; t < total;
       t += (long long)gridDim.x * blockDim.x) {
    const float* p = h + t * K;
    float m = p[0];
    for (int k = 1; k < K; ++k) m = fmaxf(m, p[k]);
    out[t] = m;
  }
}

// out = h*(1 + st + sc) + sht + shc + x  (per-channel scale/shift, split halves)
__global__ void k_modulate_res(const float* __restrict__ h, const float* __restrict__ x,
                               const float* __restrict__ sst, const float* __restrict__ ssc,
                               int C, long long Np, long long total, float* __restrict__ out)
{
  for (long long t = blockIdx.x * (long long)blockDim.x + threadIdx.x; t < total;
       t += (long long)gridDim.x * blockDim.x) {
    long long r = t / Np;
    const int c = (int)(r % C); const int b = (int)(r / C);
    const float scale = sst[(long long)b * 2 * C + c]     + ssc[(long long)b * 2 * C + c];
    const float shift = sst[(long long)b * 2 * C + C + c] + ssc[(long long)b * 2 * C + C + c];
    out[t] = h[t] * (1.f + scale) + shift + x[t];
  }
}

// For each point in xyz1, 3 nearest points in xyz2 + inverse-distance weights.
__global__ void k_knn3(const float* __restrict__ xyz1, const float* __restrict__ xyz2,
                       int N, int S, int* __restrict__ idx3, float* __restrict__ w3)
{
  const int b = blockIdx.y;
  const int n = blockIdx.x * blockDim.x + threadIdx.x;
  if (n >= N) return;
  const float px = xyz1[((long long)b * 3 + 0) * N + n];
  const float py = xyz1[((long long)b * 3 + 1) * N + n];
  const float pz = xyz1[((long long)b * 3 + 2) * N + n];
  float d0 = 3.4e38f, d1 = 3.4e38f, d2 = 3.4e38f; int i0 = 0, i1 = 0, i2 = 0;
  for (int s = 0; s < S; ++s) {
    const float dx = px - xyz2[((long long)b * 3 + 0) * S + s];
    const float dy = py - xyz2[((long long)b * 3 + 1) * S + s];
    const float dz = pz - xyz2[((long long)b * 3 + 2) * S + s];
    const float dd = dx * dx + dy * dy + dz * dz;
    if (dd < d0)      { d2 = d1; i2 = i1; d1 = d0; i1 = i0; d0 = dd; i0 = s; }
    else if (dd < d1) { d2 = d1; i2 = i1; d1 = dd; i1 = s; }
    else if (dd < d2) { d2 = dd; i2 = s; }
  }
  float w0 = 1.f / (d0 + 1e-8f), w1 = 1.f / (d1 + 1e-8f), w2 = 1.f / (d2 + 1e-8f);
  const float ws = w0 + w1 + w2;
  w0 /= ws; w1 /= ws; w2 /= ws;
  const long long o = ((long long)b * N + n) * 3;
  idx3[o] = i0; idx3[o + 1] = i1; idx3[o + 2] = i2;
  w3[o] = w0; w3[o + 1] = w1; w3[o + 2] = w2;
}

// dst[b][coff+c][n] = sum_j w3 * pts2[b][c][idx3_j]
__global__ void k_interp3(const float* __restrict__ pts2, const int* __restrict__ idx3,
                          const float* __restrict__ w3, int C2, int S, long long N,
                          float* __restrict__ dst, int Cdst, int coff, long long total)
{
  for (long long t = blockIdx.x * (long long)blockDim.x + threadIdx.x; t < total;
       t += (long long)gridDim.x * blockDim.x) {
    const long long n = t % N; long long r = t / N;
    const int c = (int)(r % C2); const int b = (int)(r / C2);
    const long long o = ((long long)b * N + n) * 3;
    const float* p = pts2 + ((long long)b * C2 + c) * S;
    const float v = w3[o] * p[idx3[o]] + w3[o + 1] * p[idx3[o + 1]] + w3[o + 2] * p[idx3[o + 2]];
    dst[((long long)b * Cdst + coff + c) * N + n] = v;
  }
}

// dst[b][coff+c][n] = src[b][c][n]
__global__ void k_copy_channels(const float* __restrict__ src, float* __restrict__ dst,
                                int Csrc, int Cdst, int coff, long long N, long long total)
{
  for (long long t = blockIdx.x * (long long)blockDim.x + threadIdx.x; t < total;
       t += (long long)gridDim.x * blockDim.x) {
    const long long n = t % N; long long r = t / N;
    const int c = (int)(r % Csrc); const int b = (int)(r / Csrc);
    dst[((long long)b * Cdst + coff + c) * N + n] = src[t];
  }
}

// final conv (C -> 3) fused with transpose to out[B][N][3]
__global__ void k_final(const float* __restrict__ W, const float* __restrict__ bias,
                        const float* __restrict__ feat, float* __restrict__ out, int N, int C)
{
  const int b = blockIdx.y;
  const int n = blockIdx.x * blockDim.x + threadIdx.x;
  if (n >= N) return;
  for (int o = 0; o < 3; ++o) {
    float s = bias[o];
    const float* w = W + (long long)o * C;
    for (int c = 0; c < C; ++c) s += w[c] * feat[((long long)b * C + c) * N + n];
    out[((long long)b * N + n) * 3 + o] = s;
  }
}

// ============================ host-side plumbing ============================

struct LinP { const float* b; const float* w; };
struct GNP  { const float* b; const float* g; };
struct BlkP { LinP c1, c2, cp; GNP g1, g2; LinP tp; };
struct SAP  { BlkP blk; GNP gn; LinP mlp; };   // same key layout as fp: {blk, gn, mlp}

struct Arena {
  char* base; size_t cur;
  static size_t au(size_t x) { return (x + 255) & ~(size_t)255; }
  float* f(size_t n) { float* p = (float*)(base + cur); cur += au(n * sizeof(float)); return p; }
  int*   i(size_t n) { int*   p = (int*)(base + cur);   cur += au(n * sizeof(int));   return p; }
};

static inline dim3 g1(long long total) {
  long long bl = (total + THREADS - 1) / THREADS;
  if (bl > (1 << 20)) bl = (1 << 20);
  if (bl < 1) bl = 1;
  return dim3((unsigned)bl);
}

static void run_gemm(hipStream_t st, const float* W, const float* bias, const float* X, float* Y,
                     int M, int K, int N, int B, long long xs, long long ys)
{
  // 128 threads = 4 waves; each wave covers 32 output columns.
  dim3 grid((N + 127) / 128, (M + 15) / 16, B);
  k_gemm_wmma<<<grid, 128, 0, st>>>(W, bias, X, Y, M, K, N, xs, ys);
}

static void run_gn(hipStream_t st, float* x, const GNP& p, int C, long long Np, int B, int doSilu)
{
  dim3 grid(B, 8);  // G = 8
  k_groupnorm<<<grid, THREADS, 0, st>>>(x, p.g, p.b, C, Np, 8, doSilu);
}

static void run_small(hipStream_t st, const LinP& l, const float* in, float* out,
                      int I, int O, int B, int doSilu)
{
  dim3 grid((O + THREADS - 1) / THREADS, B);
  k_small_linear<<<grid, THREADS, 0, st>>>(l.w, l.b, in, out, I, O, doSilu);
}

static void run_invres(hipStream_t st, const BlkP& p, const float* x, float* out,
                       int C, long long Np, int B, const float* temb, const float* cemb, Arena& A)
{
  float* h1  = A.f((size_t)B * 4 * C * Np);
  float* h2  = A.f((size_t)B * C * Np);
  float* sst = A.f((size_t)B * 2 * C);
  float* ssc = A.f((size_t)B * 2 * C);
  run_gemm(st, p.c1.w, p.c1.b, x,  h1, 4 * C, C,     (int)Np, B, (long long)C * Np,     (long long)4 * C * Np);
  run_gn(st, h1, p.g1, 4 * C, Np, B, 1);
  run_gemm(st, p.c2.w, p.c2.b, h1, h2, C,     4 * C, (int)Np, B, (long long)4 * C * Np, (long long)C * Np);
  run_gn(st, h2, p.g2, C, Np, B, 0);
  run_small(st, p.tp, temb, sst, C, 2 * C, B, 0);
  run_small(st, p.cp, cemb, ssc, C, 2 * C, B, 0);
  const long long total = (long long)B * C * Np;
  k_modulate_res<<<g1(total), THREADS, 0, st>>>(h2, x, sst, ssc, C, Np, total, out);
}

static void run_sa(hipStream_t st, const SAP& p, int npoint, float radius, int K,
                   const float* xyz, const float* feat, int Cin, int Nin, int Cout,
                   float* new_xyz, float* outFeat,
                   const float* temb, const float* cemb, int B, Arena& A)
{
  // In this model Nin > npoint at every level, so always FPS-sample.
  float* dist = A.f((size_t)B * Nin);
  int*   fidx = A.i((size_t)B * npoint);
  k_fps<<<B, THREADS, 0, st>>>(xyz, Nin, npoint, fidx, dist);
  {
    dim3 grid((npoint + THREADS - 1) / THREADS, 3, B);
    k_gather_xyz<<<grid, THREADS, 0, st>>>(xyz, fidx, Nin, npoint, new_xyz);
  }
  const int S = npoint;
  int* gidx = A.i((size_t)B * S * K);
  {
    dim3 grid((S + 127) / 128, B);
    k_query_ball<<<grid, 128, 0, st>>>(xyz, new_xyz, Nin, S, radius * radius, K, gidx);
  }
  const int Cg = Cin + 3;
  float* g = A.f((size_t)B * Cg * S * K);
  {
    const long long total = (long long)B * Cg * S * K;
    k_build_group<<<g1(total), THREADS, 0, st>>>(xyz, new_xyz, feat, gidx, Nin, S, K, Cin, total, g);
  }
  float* hg = A.f((size_t)B * Cout * S * K);
  run_gemm(st, p.mlp.w, p.mlp.b, g, hg, Cout, Cg, S * K, B,
           (long long)Cg * S * K, (long long)Cout * S * K);
  run_gn(st, hg, p.gn, Cout, (long long)S * K, B, 1);
  float* pooled = A.f((size_t)B * Cout * S);
  {
    const long long total = (long long)B * Cout * S;
    k_maxpool<<<g1(total), THREADS, 0, st>>>(hg, pooled, K, total);
  }
  run_invres(st, p.blk, pooled, outFeat, Cout, S, B, temb, cemb, A);
}

static void run_fp(hipStream_t st, const SAP& p,
                   const float* xyz1, const float* pts1, int C1, int N,
                   const float* xyz2, const float* pts2, int C2, int S,
                   int Cout, float* outFeat,
                   const float* temb, const float* cemb, int B, Arena& A)
{
  int*   idx3 = A.i((size_t)B * N * 3);
  float* w3   = A.f((size_t)B * N * 3);
  {
    dim3 grid((N + 127) / 128, B);
    k_knn3<<<grid, 128, 0, st>>>(xyz1, xyz2, N, S, idx3, w3);
  }
  const int Cin = C1 + C2;
  float* nfb = A.f((size_t)B * Cin * N);
  if (pts1) {
    const long long total = (long long)B * C1 * N;
    k_copy_channels<<<g1(total), THREADS, 0, st>>>(pts1, nfb, C1, Cin, 0, (long long)N, total);
  }
  {
    const long long total = (long long)B * C2 * N;
    k_interp3<<<g1(total), THREADS, 0, st>>>(pts2, idx3, w3, C2, S, (long long)N, nfb, Cin, C1, total);
  }
  float* h = A.f((size_t)B * Cout * N);
  run_gemm(st, p.mlp.w, p.mlp.b, nfb, h, Cout, Cin, N, B,
           (long long)Cin * N, (long long)Cout * N);
  run_gn(st, h, p.gn, Cout, N, B, 1);
  run_invres(st, p.blk, h, outFeat, Cout, N, B, temb, cemb, A);
}

extern "C" void kernel_launch(void* const* d_in, const int* in_sizes, int n_in,
                              void* d_out, int out_size, void* d_ws, size_t ws_size,
                              hipStream_t stream)
{
  (void)n_in; (void)out_size; (void)ws_size;
  const float* x    = (const float*)d_in[0];
  const float* t    = (const float*)d_in[1];
  const int*   sidx = (const int*)d_in[2];
  const int*   tidx = (const int*)d_in[3];
  const float* hn   = (const float*)d_in[4];
  const int B  = in_sizes[1];
  const int N0 = in_sizes[0] / (3 * B);
  const int Dm = 64;

  // ---- params flattened in jax pytree order (dict keys sorted) starting at d_in[5] ----
  int pi = 5;
  auto nf     = [&]() { return (const float*)d_in[pi++]; };
  auto getLin = [&]() { LinP l; l.b = nf(); l.w = nf(); return l; };            // {'b','w'}
  auto getGN  = [&]() { GNP g; g.b = nf(); g.g = nf(); return g; };             // {'b','g'}
  auto getBlk = [&]() { BlkP k; k.c1 = getLin(); k.c2 = getLin(); k.cp = getLin();
                        k.g1 = getGN(); k.g2 = getGN(); k.tp = getLin(); return k; }; // c1,c2,cp,g1,g2,tp
  auto getSA  = [&]() { SAP s; s.blk = getBlk(); s.gn = getGN(); s.mlp = getLin(); return s; }; // blk,gn,mlp

  LinP cproj[4]; for (int i = 0; i < 4; ++i) cproj[i] = getLin();               // 'cproj'
  SAP fp1 = getSA(), fp2 = getSA(), fp3 = getSA(), fp4 = getSA();               // 'fp1'..'fp4'
  LinP finalL = getLin();                                                       // 'final'
  LinP h1L = getLin(), h2L = getLin();                                          // 'h_mlp1','h_mlp2'
  SAP sa1 = getSA(), sa2 = getSA(), sa3 = getSA(), sa4 = getSA();               // 'sa1'..'sa4'
  const float* species_emb = nf();                                              // 'species_emb'
  LinP t1L = getLin(), t2L = getLin();                                          // 't_mlp1','t_mlp2'
  LinP tproj[4]; for (int i = 0; i < 4; ++i) tproj[i] = getLin();               // 'tproj'
  const float* type_emb = nf();                                                 // 'type_emb'

  // ---- workspace layout: persistent tensors, then reusable per-stage arena ----
  Arena A{ (char*)d_ws, 0 };
  float* xyz0     = A.f((size_t)B * 3 * N0);
  float* temb_raw = A.f((size_t)B * 256);
  float* th = A.f((size_t)B * Dm);
  float* tg = A.f((size_t)B * Dm);
  float* hh = A.f((size_t)B * Dm);
  float* hp = A.f((size_t)B * Dm);
  float* cg = A.f((size_t)B * Dm);
  const int dims[4] = {64, 128, 256, 512};
  float* tembL[4]; float* cembL[4];
  for (int i = 0; i < 4; ++i) { tembL[i] = A.f((size_t)B * dims[i]); cembL[i] = A.f((size_t)B * dims[i]); }
  float* l1x = A.f((size_t)B * 3 * 2048); float* l1p = A.f((size_t)B * 64  * 2048);
  float* l2x = A.f((size_t)B * 3 * 512);  float* l2p = A.f((size_t)B * 128 * 512);
  float* l3x = A.f((size_t)B * 3 * 128);  float* l3p = A.f((size_t)B * 256 * 128);
  float* l4x = A.f((size_t)B * 3 * 32);   float* l4p = A.f((size_t)B * 512 * 32);
  float* l3q = A.f((size_t)B * 256 * 128);
  float* l2q = A.f((size_t)B * 128 * 512);
  float* l1q = A.f((size_t)B * 64  * 2048);
  float* l0p = A.f((size_t)B * 64  * N0);
  const size_t mark = A.cur;

  // ---- embeddings ----
  { dim3 grid(1, B); k_timestep_embed<<<grid, 256, 0, stream>>>(t, temb_raw, 128); }
  run_small(stream, t1L, temb_raw, th, 256, Dm, B, 1);
  run_small(stream, t2L, th, tg, Dm, Dm, B, 0);
  run_small(stream, h1L, hn, hh, 1, Dm, B, 1);
  run_small(stream, h2L, hh, hp, Dm, Dm, B, 0);
  { dim3 grid((Dm + THREADS - 1) / THREADS, B);
    k_embed_sum<<<grid, THREADS, 0, stream>>>(species_emb, type_emb, sidx, tidx, hp, cg, Dm); }
  for (int i = 0; i < 4; ++i) {
    run_small(stream, tproj[i], tg, tembL[i], Dm, dims[i], B, 0);
    run_small(stream, cproj[i], cg, cembL[i], Dm, dims[i], B, 0);
  }

  { dim3 grid((N0 + THREADS - 1) / THREADS, 3, B);
    k_transpose_xyz<<<grid, THREADS, 0, stream>>>(x, xyz0, N0); }

  // ---- encoder ----
  A.cur = mark; run_sa(stream, sa1, 2048, 0.1f, 32, xyz0, nullptr, 0,   N0,   64,  l1x, l1p, tembL[0], cembL[0], B, A);
  A.cur = mark; run_sa(stream, sa2, 512,  0.2f, 32, l1x,  l1p,    64,  2048, 128, l2x, l2p, tembL[1], cembL[1], B, A);
  A.cur = mark; run_sa(stream, sa3, 128,  0.4f, 32, l2x,  l2p,    128, 512,  256, l3x, l3p, tembL[2], cembL[2], B, A);
  A.cur = mark; run_sa(stream, sa4, 32,   0.8f, 32, l3x,  l3p,    256, 128,  512, l4x, l4p, tembL[3], cembL[3], B, A);

  // ---- decoder ----
  A.cur = mark; run_fp(stream, fp4, l3x, l3p, 256, 128,  l4x, l4p, 512, 32,  256, l3q, tembL[2], cembL[2], B, A);
  A.cur = mark; run_fp(stream, fp3, l2x, l2p, 128, 512,  l3x, l3q, 256, 128, 128, l2q, tembL[1], cembL[1], B, A);
  A.cur = mark; run_fp(stream, fp2, l1x, l1p, 64,  2048, l2x, l2q, 128, 512, 64,  l1q, tembL[0], cembL[0], B, A);
  A.cur = mark; run_fp(stream, fp1, xyz0, nullptr, 0, N0, l1x, l1q, 64, 2048, 64, l0p, tembL[0], cembL[0], B, A);

  // ---- head ----
  { dim3 grid((N0 + 127) / 128, B);
    k_final<<<grid, 128, 0, stream>>>(finalL.w, finalL.b, l0p, (float*)d_out, N0, 64); }
}